// S4DSeq2Val_4930622456182
// MI455X (gfx1250) — compile-verified
//
#include <hip/hip_runtime.h>

typedef __attribute__((ext_vector_type(2))) float v2f;
typedef __attribute__((ext_vector_type(8))) float v8f;

#define Hh 128
#define Ll 2048
#define Nn 64
#define Bb 32
#define HLc (Hh * Ll)

// ---------------------------------------------------------------------------
// Kernel 1: one block per h (1024 threads). Builds the fused weight vector
//   w[h*L + j] = (W_out[h]/L) * (KC[h, L-1-j] + D[h])
// where K[h,l] = 2*Re sum_n Cd[h,n] exp(dtA[h,n] l), KC = inclusive prefix sum.
// ---------------------------------------------------------------------------
__global__ __launch_bounds__(1024)
void s4d_build_weight(const float* __restrict__ logAr,
                      const float* __restrict__ Aim,
                      const float* __restrict__ Cre,
                      const float* __restrict__ Cim,
                      const float* __restrict__ logdt,
                      const float* __restrict__ Dsk,
                      const float* __restrict__ Wout,
                      float* __restrict__ w)
{
    __shared__ float dtAr[Nn], dtAi[Nn], CdR[Nn], CdI[Nn];
    __shared__ float Ksh[Ll];
    __shared__ float scan[1024];

    const int h = blockIdx.x;
    const int t = threadIdx.x;
    const float dt = __expf(logdt[h]);

    if (t < Nn) {
        // A = -exp(logAr) + i*Aim ; dtA = A*dt ; Cd = C * expm1(dtA) / A
        float ear = __expf(logAr[h * Nn + t]);
        float ai  = Aim[h * Nn + t];
        float dr  = -ear * dt, di = ai * dt;
        dtAr[t] = dr; dtAi[t] = di;
        float ed = __expf(dr);
        float s, c; __sincosf(di, &s, &c);
        float Er = ed * c - 1.0f, Ei = ed * s;          // expm1(dtA)
        float cr = Cre[h * Nn + t], ci = Cim[h * Nn + t];
        float NumR = cr * Er - ci * Ei;
        float NumI = cr * Ei + ci * Er;
        float inv  = 1.0f / (ear * ear + ai * ai);       // |A|^-2
        CdR[t] = (-NumR * ear + NumI * ai) * inv;        // Num * conj(A)/|A|^2
        CdI[t] = (-NumR * ai - NumI * ear) * inv;
    }
    __syncthreads();

    // K at l = 2t and 2t+1 (64 modes each; sincosf for accurate range reduction)
    float k0 = 0.0f, k1 = 0.0f;
    const float l0 = (float)(2 * t), l1 = (float)(2 * t + 1);
    #pragma unroll 4
    for (int n = 0; n < Nn; ++n) {
        float dr = dtAr[n], di = dtAi[n], a = CdR[n], bq = CdI[n];
        float e0 = __expf(dr * l0), e1 = __expf(dr * l1);
        float s0, c0, s1, c1;
        sincosf(di * l0, &s0, &c0);
        sincosf(di * l1, &s1, &c1);
        k0 += e0 * (a * c0 - bq * s0);
        k1 += e1 * (a * c1 - bq * s1);
    }
    k0 *= 2.0f; k1 *= 2.0f;
    Ksh[2 * t] = k0; Ksh[2 * t + 1] = k1;

    // inclusive scan of the 1024 pair-sums (Hillis-Steele in LDS)
    scan[t] = k0 + k1;
    __syncthreads();
    for (int off = 1; off < 1024; off <<= 1) {
        float v = (t >= off) ? scan[t - off] : 0.0f;
        __syncthreads();
        scan[t] += v;
        __syncthreads();
    }
    float kc1 = scan[t];        // KC[2t+1] inclusive
    float kc0 = kc1 - k1;       // KC[2t]

    float Dh = Dsk[h];
    float sc = Wout[h] * (1.0f / (float)Ll);
    // j = L-1-l (reversed write)
    w[h * Ll + (Ll - 1 - 2 * t)]       = sc * (kc0 + Dh);
    w[h * Ll + (Ll - 1 - (2 * t + 1))] = sc * (kc1 + Dh);
}

// ---------------------------------------------------------------------------
// Kernel 2: bandwidth-bound matvec  partial[blk][b] = sum_t x[b, t]*w[t]
// via V_WMMA_F32_16X16X4_F32.  8 waves: wave = (slice<<1)|batch_group.
//   A (16x4 f32): lane m=lane&15 -> batch; lanes<16 hold K=0,1; lanes>=16 K=2,3
//   B (4x16 f32): w chunk broadcast across all 16 N columns (same K mapping)
// D columns identical -> lane 0 carries 8 batch partials, lane 16 the other 8.
// Inner loop: 16 b64 loads issued up front (immediate offsets), then 8 WMMAs
// on TWO independent accumulator chains (pipelined matrix unit -> 2x ILP,
// no back-to-back C-operand RAW stalls), uniform trip count.
// ---------------------------------------------------------------------------
__global__ __launch_bounds__(256)
void s4d_dot(const float* __restrict__ x,
             const float* __restrict__ w,
             float* __restrict__ partial)
{
    __shared__ float red[8][16];

    const int lane  = threadIdx.x & 31;
    const int wv    = threadIdx.x >> 5;
    const int grp   = wv & 1;              // batches [0,16) or [16,32)
    const int slice = wv >> 1;             // 0..3 -> 256-wide j-slice
    const int m     = lane & 15;
    const int b     = grp * 16 + m;
    const int khalf = (lane >> 4) * 2;     // 0 for lanes 0-15, 2 for 16-31
    const int tBase = blockIdx.x * 1024 + slice * 256;

    const float* pa = x + (size_t)b * HLc + tBase + khalf;
    const float* pb = w + tBase + khalf;

    v8f acc0 = {0.f, 0.f, 0.f, 0.f, 0.f, 0.f, 0.f, 0.f};
    v8f acc1 = {0.f, 0.f, 0.f, 0.f, 0.f, 0.f, 0.f, 0.f};

    #pragma unroll 1
    for (int i = 0; i < 8; ++i) {          // 8 x 32 t-values = 256 per wave
        v2f av[8], bv[8];
        #pragma unroll
        for (int u = 0; u < 8; ++u) {      // 16 loads in flight
            av[u] = *(const v2f*)(pa + 4 * u);
            bv[u] = *(const v2f*)(pb + 4 * u);
        }
        __builtin_prefetch(pa + 128, 0, 3);   // stream ahead
        #pragma unroll
        for (int u = 0; u < 8; u += 2) {      // two independent C chains
            acc0 = __builtin_amdgcn_wmma_f32_16x16x4_f32(
                       false, av[u],     false, bv[u],     (short)0, acc0, false, false);
            acc1 = __builtin_amdgcn_wmma_f32_16x16x4_f32(
                       false, av[u + 1], false, bv[u + 1], (short)0, acc1, false, false);
        }
        pa += 32; pb += 32;
    }

    v8f acc;
    #pragma unroll
    for (int i = 0; i < 8; ++i) acc[i] = acc0[i] + acc1[i];

    if (lane == 0) {
        #pragma unroll
        for (int i = 0; i < 8; ++i) red[wv][i] = acc[i];        // batches grp*16+0..7
    }
    if (lane == 16) {
        #pragma unroll
        for (int i = 0; i < 8; ++i) red[wv][8 + i] = acc[i];    // batches grp*16+8..15
    }
    __syncthreads();

    if (threadIdx.x < Bb) {
        int bb  = threadIdx.x;
        int g   = bb >> 4;          // batch group
        int idx = bb & 15;
        float s = red[g][idx] + red[g + 2][idx] + red[g + 4][idx] + red[g + 6][idx];
        partial[blockIdx.x * Bb + bb] = s;
    }
}

// ---------------------------------------------------------------------------
// Kernel 3: out[b] = sum_g partial[g][b] + b_out
// ---------------------------------------------------------------------------
__global__ void s4d_finalize(const float* __restrict__ partial,
                             const float* __restrict__ bout,
                             float* __restrict__ out, int G)
{
    int b = threadIdx.x;
    if (b < Bb) {
        float s = 0.0f;
        for (int g = 0; g < G; ++g) s += partial[(size_t)g * Bb + b];
        out[b] = s + bout[0];
    }
}

extern "C" void kernel_launch(void* const* d_in, const int* in_sizes, int n_in,
                              void* d_out, int out_size, void* d_ws, size_t ws_size,
                              hipStream_t stream)
{
    (void)in_sizes; (void)n_in; (void)out_size; (void)ws_size;
    const float* x      = (const float*)d_in[0];
    const float* logAr  = (const float*)d_in[1];
    const float* Aim    = (const float*)d_in[2];
    const float* Cre    = (const float*)d_in[3];
    const float* Cim    = (const float*)d_in[4];
    const float* logdt  = (const float*)d_in[5];
    const float* Dsk    = (const float*)d_in[6];
    const float* Wout   = (const float*)d_in[7];
    const float* bout   = (const float*)d_in[8];
    float* out = (float*)d_out;

    float* w       = (float*)d_ws;           // H*L floats = 1 MB
    float* partial = w + HLc;                // 256*32 floats

    const int G = 256;                        // HL / 1024

    s4d_build_weight<<<dim3(Hh), dim3(1024), 0, stream>>>(
        logAr, Aim, Cre, Cim, logdt, Dsk, Wout, w);

    s4d_dot<<<dim3(G), dim3(256), 0, stream>>>(x, w, partial);

    s4d_finalize<<<dim3(1), dim3(32), 0, stream>>>(partial, bout, out, G);
}